// Reinforce_55662776156760
// MI455X (gfx1250) — compile-verified
//
#include <hip/hip_runtime.h>
#include <hip/hip_bf16.h>
#include <stdint.h>

// Problem shape (fixed by the reference)
#define B_DIM 16
#define L_DIM 512
#define V_DIM 32000
#define BL_DIM (B_DIM * L_DIM)                     // 8192
#define N4_TOT (((long long)BL_DIM * V_DIM) / 4)   // 65,536,000 float4

#define ENT_BLOCKS 4096
#define ENT_THREADS 256
#define PIPE_DEPTH 4  // async stages: up to 3 loads in flight while computing

#define LN2F 0.6931471805599453f
#define VALUE_CONST 0.5f
#define ENTROPIE_CONST 0.005f

// ---------------- CDNA5 async global->LDS path (guarded) ----------------
#if defined(__gfx1250__) && __has_builtin(__builtin_amdgcn_global_load_async_to_lds_b128)
#define HAVE_ASYNC128 1
#else
#define HAVE_ASYNC128 0
#endif

#if HAVE_ASYNC128
typedef int v4i __attribute__((vector_size(16)));
typedef __attribute__((address_space(1))) v4i* g_v4i_ptr;  // global (AS1, printed __device__*)
typedef __attribute__((address_space(3))) v4i* l_v4i_ptr;  // LDS (AS3)
#endif

template <int N>
__device__ __forceinline__ void wait_asynccnt() {
#if defined(__gfx1250__)
#if __has_builtin(__builtin_amdgcn_s_wait_asynccnt)
  __builtin_amdgcn_s_wait_asynccnt((short)N);
#else
  asm volatile("s_wait_asynccnt %0" ::"n"(N) : "memory");
#endif
  asm volatile("" ::: "memory");  // keep LDS reads below the wait
#endif
}

__device__ __forceinline__ void async_copy16(const void* g, void* l) {
#if HAVE_ASYNC128
  // GLOBAL_LOAD_ASYNC_TO_LDS_B128: per-lane 16B global -> LDS, tracked by ASYNCcnt
  __builtin_amdgcn_global_load_async_to_lds_b128(
      (g_v4i_ptr)(uintptr_t)g,
      (l_v4i_ptr)(uint32_t)(uintptr_t)l,
      0, 0);
#else
  (void)g;
  (void)l;
#endif
}

// ---------------- reductions ----------------
__device__ __forceinline__ float wave_sum(float v) {
  // wave32 on gfx1250
  for (int off = 16; off > 0; off >>= 1) v += __shfl_xor(v, off, 32);
  return v;
}

__device__ __forceinline__ float ent_term(float4 v) {
  // x * log2(x); the ln2 factor is folded into the final scale
  return (v.x * __builtin_amdgcn_logf(v.x) + v.y * __builtin_amdgcn_logf(v.y)) +
         (v.z * __builtin_amdgcn_logf(v.z) + v.w * __builtin_amdgcn_logf(v.w));
}

// ---------------- kernels ----------------
__global__ void init_acc_kernel(float* acc) { acc[0] = 0.0f; }

__global__ __launch_bounds__(ENT_THREADS) void entropy_sum_kernel(
    const float* __restrict__ pred, float* __restrict__ acc) {
  const float4* __restrict__ src = (const float4*)pred;
  const long long n4 = N4_TOT;
  const long long stride = (long long)gridDim.x * ENT_THREADS;
  long long idx = (long long)blockIdx.x * ENT_THREADS + threadIdx.x;
  const int tid = threadIdx.x;
  float a = 0.0f;

#if HAVE_ASYNC128
  __shared__ float4 buf[PIPE_DEPTH][ENT_THREADS];  // 16 KB, 4-stage pipeline

  // Prologue: fill up to DEPTH-1 stages ahead.
  bool have = idx < n4;
  if (have) async_copy16(&src[idx], &buf[0][tid]);
  if (idx + stride < n4) async_copy16(&src[idx + stride], &buf[1][tid]);
  if (idx + 2 * stride < n4) async_copy16(&src[idx + 2 * stride], &buf[2][tid]);

  int cur = 0;
  while (have) {
    const bool v1 = idx + stride < n4;
    const bool v2 = idx + 2 * stride < n4;
    const bool v3 = idx + 3 * stride < n4;
    if (v3) {
      async_copy16(&src[idx + 3 * stride], &buf[(cur + 3) & (PIPE_DEPTH - 1)][tid]);
      wait_asynccnt<3>();  // 4 outstanding -> oldest (cur) has landed
    } else if (v2) {
      wait_asynccnt<2>();  // 3 outstanding -> oldest done
    } else if (v1) {
      wait_asynccnt<1>();  // 2 outstanding -> oldest done
    } else {
      wait_asynccnt<0>();  // last tile
    }
    a += ent_term(buf[cur][tid]);  // each lane reads only its own slot: no barrier
    idx += stride;
    have = v1;
    cur = (cur + 1) & (PIPE_DEPTH - 1);
  }
#else
  for (; idx < n4; idx += stride) a += ent_term(src[idx]);
#endif

  // block reduction: wave shuffles + LDS
  __shared__ float wsum[ENT_THREADS / 32];
  const float w = wave_sum(a);
  if ((tid & 31) == 0) wsum[tid >> 5] = w;
  __syncthreads();
  if (tid == 0) {
    float t = 0.0f;
#pragma unroll
    for (int i = 0; i < ENT_THREADS / 32; ++i) t += wsum[i];
    // entropy_loss contribution: 0.005 * (1/8192) * sum(x * ln x); log was base-2
    atomicAdd(acc, t * (ENTROPIE_CONST * LN2F / (float)BL_DIM));
  }
}

__global__ __launch_bounds__(L_DIM) void policy_value_kernel(
    const float* __restrict__ action_probs, const float* __restrict__ value,
    const float* __restrict__ critic_value, const int* __restrict__ segments,
    float* __restrict__ acc) {
  const int b = blockIdx.x;
  const int t = threadIdx.x;
  const int i = b * L_DIM + t;

  __shared__ int s[L_DIM];
  __shared__ float sums[L_DIM + 1];
  __shared__ float wsum[L_DIM / 32];

  const int seg = segments[i];
  const float lp = __builtin_amdgcn_logf(action_probs[i]) * LN2F;  // ln(p)

  s[t] = seg;
  sums[t] = 0.0f;
  if (t == 0) sums[L_DIM] = 0.0f;
  __syncthreads();

  // inclusive scan of segment markers (Hillis-Steele)
  for (int off = 1; off < L_DIM; off <<= 1) {
    const int add = (t >= off) ? s[t - off] : 0;
    __syncthreads();
    s[t] += add;
    __syncthreads();
  }
  const int incl = s[t];
  const int seg_id = incl - seg;   // exclusive scan
  const int n_seg = s[L_DIM - 1];  // markers in this row

  atomicAdd(&sums[seg_id], lp);  // ds_add_f32 segment sums
  __syncthreads();

  float seg_prob = sums[seg_id];
  // faithful quirk: only the LAST batch row's trailing tail stays zero
  if (b == B_DIM - 1 && seg_id == n_seg) seg_prob = 0.0f;

  const float adv = value[i] - critic_value[i];
  const float contrib = -adv * seg_prob + VALUE_CONST * adv * adv;

  const float w = wave_sum(contrib);
  if ((t & 31) == 0) wsum[t >> 5] = w;
  __syncthreads();
  if (t == 0) {
    float tt = 0.0f;
#pragma unroll
    for (int k = 0; k < L_DIM / 32; ++k) tt += wsum[k];
    atomicAdd(acc, tt * (1.0f / (float)BL_DIM));
  }
}

__global__ void finalize_kernel(const float* __restrict__ acc, float* __restrict__ out) {
  out[0] = acc[0];
}

// ---------------- host entry ----------------
extern "C" void kernel_launch(void* const* d_in, const int* in_sizes, int n_in,
                              void* d_out, int out_size, void* d_ws, size_t ws_size,
                              hipStream_t stream) {
  (void)in_sizes; (void)n_in; (void)out_size; (void)ws_size;
  const float* pred = (const float*)d_in[0];
  const float* action_probs = (const float*)d_in[1];
  const float* value = (const float*)d_in[2];
  const float* critic_value = (const float*)d_in[3];
  const int* segments = (const int*)d_in[4];
  float* out = (float*)d_out;
  float* acc = (float*)d_ws;

  init_acc_kernel<<<1, 1, 0, stream>>>(acc);
  entropy_sum_kernel<<<ENT_BLOCKS, ENT_THREADS, 0, stream>>>(pred, acc);
  policy_value_kernel<<<B_DIM, L_DIM, 0, stream>>>(action_probs, value, critic_value,
                                                   segments, acc);
  finalize_kernel<<<1, 1, 0, stream>>>(acc, out);
}